// QuantumLayer_81518479278603
// MI455X (gfx1250) — compile-verified
//
#include <hip/hip_runtime.h>

typedef __attribute__((ext_vector_type(2))) float v2f;
typedef __attribute__((ext_vector_type(4))) float v4f;
typedef __attribute__((ext_vector_type(8))) float v8f;

// ---------------------------------------------------------------------------
// Kernel 1: build the fixed 16x16 complex unitary of the entangling layer.
// Thread j propagates basis state |j> through Rot(phi,theta,omega) on each
// wire followed by the CNOT ring (r = 0 % 3 + 1 = 1).
// Layout in ws: U_re[16][16] (row i = output basis, col j = input basis),
// then U_im[16][16].  Qubit q owns bit (3-q) of the basis index.
// ---------------------------------------------------------------------------
__global__ void build_unitary_kernel(const float* __restrict__ w,
                                     float* __restrict__ U) {
    const int j = threadIdx.x;
    if (j >= 16) return;

    float sr[16], si[16];
#pragma unroll
    for (int i = 0; i < 16; ++i) { sr[i] = (i == j) ? 1.0f : 0.0f; si[i] = 0.0f; }

    // Rot = RZ(omega) RY(theta) RZ(phi) on each wire
#pragma unroll
    for (int q = 0; q < 4; ++q) {
        const float phi = w[q * 3 + 0];
        const float th  = w[q * 3 + 1];
        const float om  = w[q * 3 + 2];
        float ch, sh, cp, sp, cm, sm;
        __sincosf(0.5f * th, &sh, &ch);
        __sincosf(0.5f * (phi + om), &sp, &cp);
        __sincosf(0.5f * (phi - om), &sm, &cm);
        const float g00r =  cp * ch, g00i = -sp * ch;   // e^{-i(p+o)/2} cos
        const float g01r = -cm * sh, g01i = -sm * sh;   // -e^{ i(p-o)/2} sin
        const float g10r =  cm * sh, g10i = -sm * sh;   // e^{-i(p-o)/2} sin
        const float g11r =  cp * ch, g11i =  sp * ch;   // e^{ i(p+o)/2} cos
        const int bit = 8 >> q;
#pragma unroll
        for (int i0 = 0; i0 < 16; ++i0) {
            if (i0 & bit) continue;
            const int i1 = i0 | bit;
            const float ar = sr[i0], ai = si[i0];
            const float br = sr[i1], bi = si[i1];
            sr[i0] = g00r * ar - g00i * ai + g01r * br - g01i * bi;
            si[i0] = g00r * ai + g00i * ar + g01r * bi + g01i * br;
            sr[i1] = g10r * ar - g10i * ai + g11r * br - g11i * bi;
            si[i1] = g10r * ai + g10i * ar + g11r * bi + g11i * br;
        }
    }

    // CNOT ring with range 1: control q, target (q+1)%4
#pragma unroll
    for (int q = 0; q < 4; ++q) {
        const int cbit = 8 >> q;
        const int tbit = 8 >> ((q + 1) & 3);
#pragma unroll
        for (int i = 0; i < 16; ++i) {
            if ((i & cbit) && !(i & tbit)) {
                const int i2 = i | tbit;
                float t;
                t = sr[i]; sr[i] = sr[i2]; sr[i2] = t;
                t = si[i]; si[i] = si[i2]; si[i2] = t;
            }
        }
    }

#pragma unroll
    for (int i = 0; i < 16; ++i) {
        U[i * 16 + j]       = sr[i];
        U[256 + i * 16 + j] = si[i];
    }
}

// ---------------------------------------------------------------------------
// Kernel 2: main batched circuit via V_WMMA_F32_16X16X4_F32.
// Each wave: 16 samples per tile.  D_re/D_im[16x16] = U_{re,im}[16x16] x
// Psi[16x16samples], K split into 4 chunks of 4.  Then p = re^2+im^2 and the
// PauliZ +/-1 masks are folded per lane-half, combined with shfl_xor(16).
//
// A 16x4 f32 layout (ISA 7.12.2): lane L holds row M=L&15; VGPR v holds
// K = v + 2*(L>=16).  B 4x16: VGPR v holds row K = v + 2*(L>=16), col = L&15.
// C/D 16x16: VGPR j holds row M = j + 8*(L>=16), col N = L&15.
// ---------------------------------------------------------------------------
__global__ void __launch_bounds__(256) qcircuit_wmma_kernel(
    const float* __restrict__ x,   // [B,4]
    const float* __restrict__ U,   // U_re[256] then U_im[256]
    float* __restrict__ out,       // [4,B]
    int batch) {
    const int lane = threadIdx.x & 31;
    const int m    = lane & 15;    // output row / sample column within tile
    const int hi   = lane >> 4;    // lane-half selector

    // Load U into WMMA A-matrix registers once per wave (re + im, 4 K-blocks).
    v2f Are[4], Aim[4];
#pragma unroll
    for (int kb = 0; kb < 4; ++kb) {
        const float* pr = U +       m * 16 + kb * 4 + 2 * hi;
        const float* pi = U + 256 + m * 16 + kb * 4 + 2 * hi;
        Are[kb].x = pr[0]; Are[kb].y = pr[1];
        Aim[kb].x = pi[0]; Aim[kb].y = pi[1];
    }

    const int tiles  = batch >> 4;
    const int wave   = (int)((blockIdx.x * blockDim.x + threadIdx.x) >> 5);
    const int nwaves = (int)((gridDim.x * blockDim.x) >> 5);

    for (int t = wave; t < tiles; t += nwaves) {
        const int tb = t << 4;
        const int n  = tb + m;

        // Embedding: psi[i] = prod_q (bit ? sin(x_q/2) : cos(x_q/2)), real.
        const v4f xv = *(const v4f*)(x + n * 4);
        float c0, s0, c1, s1, c2, s2, c3, s3;
        __sincosf(0.5f * xv.x, &s0, &c0);
        __sincosf(0.5f * xv.y, &s1, &c1);
        __sincosf(0.5f * xv.z, &s2, &c2);
        __sincosf(0.5f * xv.w, &s3, &c3);

        // psi[i] = p01[i>>2] * p23[i&3]   (i bit3..bit0 = q0..q3)
        const float p01_0 = c0 * c1, p01_1 = c0 * s1, p01_2 = s0 * c1, p01_3 = s0 * s1;
        const float p23_0 = c2 * c3, p23_1 = c2 * s3, p23_2 = s2 * c3, p23_3 = s2 * s3;
        // This lane's B rows need k = 4*kb + v + 2*hi  =>  i&3 in {2hi, 2hi+1}
        const float q0 = hi ? p23_2 : p23_0;
        const float q1 = hi ? p23_3 : p23_1;
        const float p01[4] = {p01_0, p01_1, p01_2, p01_3};

        v8f dre = {0.f, 0.f, 0.f, 0.f, 0.f, 0.f, 0.f, 0.f};
        v8f dim = {0.f, 0.f, 0.f, 0.f, 0.f, 0.f, 0.f, 0.f};
#pragma unroll
        for (int kb = 0; kb < 4; ++kb) {
            v2f bv;
            bv.x = p01[kb] * q0;
            bv.y = p01[kb] * q1;
            dre = __builtin_amdgcn_wmma_f32_16x16x4_f32(
                false, Are[kb], false, bv, (short)0, dre, false, false);
            dim = __builtin_amdgcn_wmma_f32_16x16x4_f32(
                false, Aim[kb], false, bv, (short)0, dim, false, false);
        }

        // Probabilities for the 8 rows this lane holds (M = j + 8*hi).
        float p[8];
#pragma unroll
        for (int jj = 0; jj < 8; ++jj)
            p[jj] = dre[jj] * dre[jj] + dim[jj] * dim[jj];

        // PauliZ signs: q0 -> sign of hi; q1 -> j>>2; q2 -> (j>>1)&1; q3 -> j&1
        const float S   = ((p[0] + p[1]) + (p[2] + p[3])) + ((p[4] + p[5]) + (p[6] + p[7]));
        float s0v = hi ? -S : S;
        float s1v = (p[0] + p[1] + p[2] + p[3]) - (p[4] + p[5] + p[6] + p[7]);
        float s2v = (p[0] + p[1] + p[4] + p[5]) - (p[2] + p[3] + p[6] + p[7]);
        float s3v = (p[0] + p[2] + p[4] + p[6]) - (p[1] + p[3] + p[5] + p[7]);

        // Combine the two half-columns (rows 0-7 and 8-15) of each sample.
        s0v += __shfl_xor(s0v, 16, 32);
        s1v += __shfl_xor(s1v, 16, 32);
        s2v += __shfl_xor(s2v, 16, 32);
        s3v += __shfl_xor(s3v, 16, 32);

        // out is [4, B]; low half stores q=0,1, high half stores q=2,3.
        if (hi == 0) {
            out[tb + m]         = s0v;
            out[batch + tb + m] = s1v;
        } else {
            out[2 * batch + tb + m] = s2v;
            out[3 * batch + tb + m] = s3v;
        }
    }
}

// ---------------------------------------------------------------------------
extern "C" void kernel_launch(void* const* d_in, const int* in_sizes, int n_in,
                              void* d_out, int out_size, void* d_ws, size_t ws_size,
                              hipStream_t stream) {
    (void)n_in; (void)out_size; (void)ws_size;
    const float* x = (const float*)d_in[0];   // [B, 4] float32
    const float* w = (const float*)d_in[1];   // [1, 4, 3] float32
    float* out = (float*)d_out;               // [4, B] float32
    float* U   = (float*)d_ws;                // 512 floats scratch

    hipLaunchKernelGGL(build_unitary_kernel, dim3(1), dim3(16), 0, stream, w, U);

    const int batch = in_sizes[0] / 4;
    const int tiles = batch >> 4;
    const int threads = 256;                 // 8 waves/block
    const int wavesPerBlock = threads / 32;
    int blocks = (tiles + wavesPerBlock - 1) / wavesPerBlock;
    if (blocks > 2048) blocks = 2048;        // grid-stride over tiles
    if (blocks < 1) blocks = 1;
    hipLaunchKernelGGL(qcircuit_wmma_kernel, dim3(blocks), dim3(threads), 0,
                       stream, x, U, out, batch);
}